// LIFSpike_26242250178612
// MI455X (gfx1250) — compile-verified
//
#include <hip/hip_runtime.h>
#include <cstdint>
#include <cstddef>

// LIF spike recurrence: T=8 timesteps, tau=0.5, thresh=1.0 (forward value of the
// straight-through spike is exactly the hard spike).
//
// Memory-bound streaming kernel: 128 MiB in + 128 MiB out, ~11.5 us floor at
// 23.3 TB/s. Data path: GLOBAL_LOAD_ASYNC_TO_LDS_B128 (8 in-flight strided
// plane reads per thread, ASYNCcnt-tracked) -> LDS -> registers -> nontemporal
// b128 stores.

#define LIF_T      8
#define LIF_TAU    0.5f
#define LIF_THRESH 1.0f
#define BLOCK      256

typedef float v4f __attribute__((ext_vector_type(4)));
typedef int   gv4i __attribute__((vector_size(16)));   // matches builtin param type

#define AS1 __attribute__((address_space(1)))
#define AS3 __attribute__((address_space(3)))

__global__ __launch_bounds__(BLOCK) void lif_spike_kernel(
    const float* __restrict__ x, float* __restrict__ out, unsigned nt4) {
  // Per-thread private staging slots: smem[t][tid], 8*256*16B = 32 KB/block.
  __shared__ v4f smem[LIF_T][BLOCK];

  const unsigned tid = threadIdx.x;
  const unsigned i = blockIdx.x * BLOCK + tid;
  if (i >= nt4) return;

  const v4f* __restrict__ x4 = (const v4f*)x;
  v4f* __restrict__ o4 = (v4f*)out;

  // ---- Stage: fire all 8 timestep-plane loads asynchronously into LDS ----
#pragma unroll
  for (int t = 0; t < LIF_T; ++t) {
    const v4f* gp = x4 + (size_t)t * (size_t)nt4 + i;
#if __has_builtin(__builtin_amdgcn_global_load_async_to_lds_b128)
    __builtin_amdgcn_global_load_async_to_lds_b128(
        (AS1 gv4i*)(gv4i*)gp,
        (AS3 gv4i*)(gv4i*)&smem[t][tid],
        /*offset=*/0, /*cpol=*/0);
#else
    unsigned lds_addr = (unsigned)(uintptr_t)&smem[t][tid];
    unsigned long long ga = (unsigned long long)(uintptr_t)gp;
    asm volatile("global_load_async_to_lds_b128 %0, %1, off"
                 :
                 : "v"(lds_addr), "v"(ga)
                 : "memory");
#endif
  }

  // Drain the async copies (same-wave producer/consumer: no barrier needed).
#if __has_builtin(__builtin_amdgcn_s_wait_asynccnt)
  __builtin_amdgcn_s_wait_asynccnt(0);
#else
  asm volatile("s_wait_asynccnt 0" ::: "memory");
#endif
  asm volatile("" ::: "memory");  // keep LDS reads below the wait

  // ---- Compute: LIF recurrence fully unrolled in registers ----
  v4f mem = {0.f, 0.f, 0.f, 0.f};
#pragma unroll
  for (int t = 0; t < LIF_T; ++t) {
    v4f xv = smem[t][tid];
    mem = mem * LIF_TAU + xv;
    v4f spike;
    spike.x = (mem.x > LIF_THRESH) ? 1.0f : 0.0f;
    spike.y = (mem.y > LIF_THRESH) ? 1.0f : 0.0f;
    spike.z = (mem.z > LIF_THRESH) ? 1.0f : 0.0f;
    spike.w = (mem.w > LIF_THRESH) ? 1.0f : 0.0f;
    mem -= spike * LIF_THRESH;
    // Output is never re-read: bypass-cache store (b128, TH_STORE_NT).
    __builtin_nontemporal_store(spike, o4 + (size_t)t * (size_t)nt4 + i);
  }
}

extern "C" void kernel_launch(void* const* d_in, const int* in_sizes, int n_in,
                              void* d_out, int out_size, void* d_ws,
                              size_t ws_size, hipStream_t stream) {
  (void)n_in; (void)out_size; (void)d_ws; (void)ws_size;
  const float* x = (const float*)d_in[0];
  float* out = (float*)d_out;

  const unsigned n = (unsigned)in_sizes[0];     // 33,554,432 elements
  const unsigned nt4 = n / (LIF_T * 4);         // float4 slots per t-plane
  const unsigned blocks = (nt4 + BLOCK - 1) / BLOCK;

  lif_spike_kernel<<<blocks, BLOCK, 0, stream>>>(x, out, nt4);
}